// KerasLIFLayerSparse_11510512353281
// MI455X (gfx1250) — compile-verified
//
#include <hip/hip_runtime.h>
#include <cstdint>

// Problem constants (from reference)
#define T_STEPS 256
#define BATCH   32
#define UNITS   4096
#define IN_DIM  4096
#define S_IN    128
#define S_OUT   128

// Fast-path tiling
#define G_SPLIT  8                   // unit groups per batch -> 32*8 = 256 blocks
#define U_BLK    (UNITS / G_SPLIT)   // 512 units per block
#define THREADS_A 128                // 4 waves, 4 units (one float4) per thread
#define DEPTH    4                   // async global->LDS pipeline depth (power of 2)

#define THREADS_M 1024               // monolithic fallback: 1 block per batch

typedef __attribute__((ext_vector_type(4))) float v4f;  // clang-native vec4

// NT store: the 134 MB state stream must not evict the L2-resident 64 MB Wt.
__device__ __forceinline__ void nt_store_f4(float* p, const float4& v) {
  v4f nv;
  nv.x = v.x; nv.y = v.y; nv.z = v.z; nv.w = v.w;
  __builtin_nontemporal_store(nv, (v4f*)p);
}

// ---------------------------------------------------------------------------
// Weight transpose: Wt[j][u] = W[u][j]  (so gathered spike rows are coalesced)
// ---------------------------------------------------------------------------
__global__ __launch_bounds__(256) void wt_transpose(const float* __restrict__ W,
                                                    float* __restrict__ Wt) {
  __shared__ float tile[32][33];
  const int jx = blockIdx.x * 32 + threadIdx.x;  // column of W (input index)
  const int u0 = blockIdx.y * 32;                // row of W (unit)
#pragma unroll
  for (int i = 0; i < 32; i += 8)
    tile[threadIdx.y + i][threadIdx.x] =
        W[(size_t)(u0 + threadIdx.y + i) * IN_DIM + jx];
  __syncthreads();
  const int ux = u0 + threadIdx.x;
  const int j0 = blockIdx.x * 32;
#pragma unroll
  for (int i = 0; i < 32; i += 8)
    Wt[(size_t)(j0 + threadIdx.y + i) * UNITS + ux] =
        tile[threadIdx.x][threadIdx.y + i];
}

// ---------------------------------------------------------------------------
// Async gather helpers (gfx1250 async-to-LDS path). Each lane owns a private
// 16 B LDS slot per pipeline stage: it reads back only what it loaded, so no
// block barriers are needed in the hot loop, only per-wave asynccnt waits.
// ---------------------------------------------------------------------------
__device__ __forceinline__ void async_row_b128(const float* gp, unsigned lds_off) {
  const uint64_t ga = (uint64_t)(uintptr_t)gp;
  asm volatile("global_load_async_to_lds_b128 %0, %1, off" ::"v"(lds_off),
               "v"(ga)
               : "memory");
}

// ---------------------------------------------------------------------------
// Fast path: per-(batch, unit-group) LIF recurrence with async gather pipeline.
// ---------------------------------------------------------------------------
__global__ __launch_bounds__(THREADS_A) void lif_gather(
    const float* __restrict__ Wt, const float* __restrict__ init_state,
    const int* __restrict__ spk_ids, const int* __restrict__ nspk,
    const float* __restrict__ decay, const float* __restrict__ thresh,
    float* __restrict__ states_out, int* __restrict__ cnt_out,
    int* __restrict__ stg_ids) {
  const int tid = threadIdx.x;
  const int g = blockIdx.x % G_SPLIT;
  const int b = blockIdx.x / G_SPLIT;
  const int ub = g * U_BLK;
  const int u4 = ub + tid * 4;  // first of 4 contiguous units owned by thread

  __shared__ int s_ids[S_IN];
  __shared__ int s_n;
  __shared__ __align__(16) float s_buf[DEPTH][U_BLK];  // 4 x 2 KB row slabs
  __shared__ int s_wsum[THREADS_A / 32];

  float4 st = *(const float4*)&init_state[(size_t)b * UNITS + u4];
  const float4 dc = *(const float4*)&decay[u4];
  const float4 th = *(const float4*)&thresh[u4];

  const int lane = tid & 31, wv = tid >> 5;
  const unsigned lds_base = (unsigned)(uintptr_t)(&s_buf[0][tid * 4]);
  const float* wbase = Wt + u4;  // + id*UNITS per spike

  for (int t = 0; t < T_STEPS; ++t) {
    const int tb = t * BATCH + b;
    if (tid < S_IN) s_ids[tid] = spk_ids[(size_t)tb * S_IN + tid];
    if (tid == 0) {
      int nn = nspk[tb];
      s_n = nn < 0 ? 0 : (nn > S_IN ? S_IN : nn);
    }
    __syncthreads();
    const int n = s_n;

    // ---- prime the async pipeline ----
#pragma unroll
    for (int s = 0; s < DEPTH; ++s) {
      if (s < n)
        async_row_b128(wbase + (size_t)s_ids[s] * UNITS,
                       lds_base + (unsigned)(s * U_BLK * 4));
    }

    float4 syn = make_float4(0.f, 0.f, 0.f, 0.f);
#define CONSUME(S)                                                         \
  do {                                                                     \
    const float4 r = *(const float4*)&s_buf[(S) & (DEPTH - 1)][tid * 4];   \
    syn.x += r.x; syn.y += r.y; syn.z += r.z; syn.w += r.w;                \
  } while (0)

    if (n >= DEPTH) {
      // steady state: exactly DEPTH-1 ops in flight after each wait
      for (int s = 0; s < n - (DEPTH - 1); ++s) {
        asm volatile("s_wait_asynccnt 3" ::: "memory");
        CONSUME(s);
        const int sn = s + DEPTH;
        if (sn < n)
          async_row_b128(wbase + (size_t)s_ids[sn] * UNITS,
                         lds_base + (unsigned)((sn & (DEPTH - 1)) * U_BLK * 4));
      }
      asm volatile("s_wait_asynccnt 2" ::: "memory");
      CONSUME(n - 3);
      asm volatile("s_wait_asynccnt 1" ::: "memory");
      CONSUME(n - 2);
      asm volatile("s_wait_asynccnt 0" ::: "memory");
      CONSUME(n - 1);
    } else {
      for (int s = 0; s < n; ++s) {  // n in 1..3: wait threshold = n-1-s
        const int tail = n - 1 - s;
        if (tail == 2)      asm volatile("s_wait_asynccnt 2" ::: "memory");
        else if (tail == 1) asm volatile("s_wait_asynccnt 1" ::: "memory");
        else                asm volatile("s_wait_asynccnt 0" ::: "memory");
        CONSUME(s);
      }
    }
#undef CONSUME

    // ---- LIF update ----
    float4 ns;
    ns.x = dc.x * st.x + (1.f - dc.x) * syn.x;
    ns.y = dc.y * st.y + (1.f - dc.y) * syn.y;
    ns.z = dc.z * st.z + (1.f - dc.z) * syn.z;
    ns.w = dc.w * st.w + (1.f - dc.w) * syn.w;
    nt_store_f4(&states_out[(size_t)tb * UNITS + u4], ns);

    const int m = (ns.x > th.x ? 1 : 0) | (ns.y > th.y ? 2 : 0) |
                  (ns.z > th.z ? 4 : 0) | (ns.w > th.w ? 8 : 0);
    st.x = (m & 1) ? 0.f : ns.x;
    st.y = (m & 2) ? 0.f : ns.y;
    st.z = (m & 4) ? 0.f : ns.z;
    st.w = (m & 8) ? 0.f : ns.w;

    // ---- ordered compaction within this unit group (unit-ascending) ----
    const int c = __popc(m);
    int incl = c;
#pragma unroll
    for (int d = 1; d < 32; d <<= 1) {
      int v = __shfl_up(incl, d, 32);
      if (lane >= d) incl += v;
    }
    if (lane == 31) s_wsum[wv] = incl;
    __syncthreads();
    int woff = 0, total = 0;
#pragma unroll
    for (int w = 0; w < THREADS_A / 32; ++w) {
      const int v = s_wsum[w];
      if (w < wv) woff += v;
      total += v;
    }
    int r = woff + incl - c;  // exclusive rank of this thread's first spike
    int* dst = &stg_ids[((size_t)tb * G_SPLIT + g) * S_OUT];
    if (m & 1) { if (r < S_OUT) dst[r] = u4 + 0; r++; }
    if (m & 2) { if (r < S_OUT) dst[r] = u4 + 1; r++; }
    if (m & 4) { if (r < S_OUT) dst[r] = u4 + 2; r++; }
    if (m & 8) { if (r < S_OUT) dst[r] = u4 + 3; r++; }
    if (tid == 0) cnt_out[(size_t)tb * G_SPLIT + g] = total;
    __syncthreads();  // protects s_ids / s_wsum reuse next timestep
  }
}

// ---------------------------------------------------------------------------
// Merge: global rank = prefix over the 8 unit-group counts (groups own
// ascending unit ranges, so concatenation preserves lowest-index-first order).
// ---------------------------------------------------------------------------
__global__ __launch_bounds__(S_OUT) void lif_merge(const int* __restrict__ cnt,
                                                   const int* __restrict__ stg,
                                                   float* __restrict__ out_ids,
                                                   float* __restrict__ num_out) {
  const int tb = blockIdx.x;
  const int tid = threadIdx.x;
  __shared__ int s_p[G_SPLIT + 1];
  if (tid == 0) {
    int a = 0;
    for (int gg = 0; gg < G_SPLIT; ++gg) {
      s_p[gg] = a;
      a += cnt[(size_t)tb * G_SPLIT + gg];
    }
    s_p[G_SPLIT] = a;
  }
  __syncthreads();
  const int total = s_p[G_SPLIT];
  const int n_out = total < S_OUT ? total : S_OUT;
  float v = 0.f;
  if (tid < n_out) {
    int gI = 0;
#pragma unroll
    for (int gg = 0; gg < G_SPLIT; ++gg)
      if (tid >= s_p[gg + 1]) gI = gg + 1;
    v = (float)stg[((size_t)tb * G_SPLIT + gI) * S_OUT + (tid - s_p[gI])];
  }
  out_ids[(size_t)tb * S_OUT + tid] = v;
  if (tid == 0) num_out[tb] = (float)n_out;
}

// ---------------------------------------------------------------------------
// Fallback: one block per batch, all 4096 units, no workspace staging.
// use_wt=1 -> W is pre-transposed [IN_DIM][UNITS]; use_wt=0 -> raw weights.
// ---------------------------------------------------------------------------
__global__ __launch_bounds__(THREADS_M) void lif_monolithic(
    const float* __restrict__ W, int use_wt, const float* __restrict__ init_state,
    const int* __restrict__ spk_ids, const int* __restrict__ nspk,
    const float* __restrict__ decay, const float* __restrict__ thresh,
    float* __restrict__ out_ids, float* __restrict__ num_out,
    float* __restrict__ states_out) {
  const int tid = threadIdx.x, b = blockIdx.x;
  const int u4 = tid * 4;
  __shared__ int s_ids[S_IN];
  __shared__ int s_n;
  __shared__ int s_wsum[THREADS_M / 32];

  float4 st = *(const float4*)&init_state[(size_t)b * UNITS + u4];
  const float4 dc = *(const float4*)&decay[u4];
  const float4 th = *(const float4*)&thresh[u4];
  const int lane = tid & 31, wv = tid >> 5;

  for (int t = 0; t < T_STEPS; ++t) {
    const int tb = t * BATCH + b;
    if (tid < S_IN) s_ids[tid] = spk_ids[(size_t)tb * S_IN + tid];
    if (tid == 0) {
      int nn = nspk[tb];
      s_n = nn < 0 ? 0 : (nn > S_IN ? S_IN : nn);
    }
    __syncthreads();
    const int n = s_n;

    float4 syn = make_float4(0.f, 0.f, 0.f, 0.f);
    for (int s = 0; s < n; ++s) {
      const int id = s_ids[s];
      if (use_wt) {
        const float4 r = *(const float4*)&W[(size_t)id * UNITS + u4];
        syn.x += r.x; syn.y += r.y; syn.z += r.z; syn.w += r.w;
      } else {
        syn.x += W[(size_t)(u4 + 0) * IN_DIM + id];
        syn.y += W[(size_t)(u4 + 1) * IN_DIM + id];
        syn.z += W[(size_t)(u4 + 2) * IN_DIM + id];
        syn.w += W[(size_t)(u4 + 3) * IN_DIM + id];
      }
    }

    float4 ns;
    ns.x = dc.x * st.x + (1.f - dc.x) * syn.x;
    ns.y = dc.y * st.y + (1.f - dc.y) * syn.y;
    ns.z = dc.z * st.z + (1.f - dc.z) * syn.z;
    ns.w = dc.w * st.w + (1.f - dc.w) * syn.w;
    nt_store_f4(&states_out[(size_t)tb * UNITS + u4], ns);

    const int m = (ns.x > th.x ? 1 : 0) | (ns.y > th.y ? 2 : 0) |
                  (ns.z > th.z ? 4 : 0) | (ns.w > th.w ? 8 : 0);
    st.x = (m & 1) ? 0.f : ns.x;
    st.y = (m & 2) ? 0.f : ns.y;
    st.z = (m & 4) ? 0.f : ns.z;
    st.w = (m & 8) ? 0.f : ns.w;

    const int c = __popc(m);
    int incl = c;
#pragma unroll
    for (int d = 1; d < 32; d <<= 1) {
      int v = __shfl_up(incl, d, 32);
      if (lane >= d) incl += v;
    }
    if (lane == 31) s_wsum[wv] = incl;
    __syncthreads();
    int woff = 0, total = 0;
    for (int w = 0; w < THREADS_M / 32; ++w) {
      const int v = s_wsum[w];
      if (w < wv) woff += v;
      total += v;
    }
    int r = woff + incl - c;
    float* dst = &out_ids[(size_t)tb * S_OUT];
    if (m & 1) { if (r < S_OUT) dst[r] = (float)(u4 + 0); r++; }
    if (m & 2) { if (r < S_OUT) dst[r] = (float)(u4 + 1); r++; }
    if (m & 4) { if (r < S_OUT) dst[r] = (float)(u4 + 2); r++; }
    if (m & 8) { if (r < S_OUT) dst[r] = (float)(u4 + 3); r++; }
    if (tid < S_OUT && tid >= total) dst[tid] = 0.f;  // pad slots
    if (tid == 0) num_out[tb] = (float)(total < S_OUT ? total : S_OUT);
    __syncthreads();
  }
}

// ---------------------------------------------------------------------------
extern "C" void kernel_launch(void* const* d_in, const int* in_sizes, int n_in,
                              void* d_out, int out_size, void* d_ws,
                              size_t ws_size, hipStream_t stream) {
  const float* W = (const float*)d_in[0];
  const float* ini = (const float*)d_in[1];
  const int* ids = (const int*)d_in[2];
  const int* nsp = (const int*)d_in[3];
  const float* dec = (const float*)d_in[4];
  const float* thr = (const float*)d_in[5];
  // d_in[6] = size_sparse_out (compile-time 128)

  float* out_ids = (float*)d_out;
  float* num_out = out_ids + (size_t)T_STEPS * BATCH * S_OUT;
  float* states = num_out + (size_t)T_STEPS * BATCH;

  const size_t wtBytes = (size_t)IN_DIM * UNITS * sizeof(float);     // 64 MB
  const size_t cntBytes = (size_t)T_STEPS * BATCH * G_SPLIT * sizeof(int);
  const size_t stgBytes =
      (size_t)T_STEPS * BATCH * G_SPLIT * S_OUT * sizeof(int);       // 32 MB

  if (ws_size >= wtBytes + cntBytes + stgBytes) {
    float* Wt = (float*)d_ws;
    int* cnt = (int*)((char*)d_ws + wtBytes);
    int* stg = (int*)((char*)d_ws + wtBytes + cntBytes);
    wt_transpose<<<dim3(IN_DIM / 32, UNITS / 32), dim3(32, 8), 0, stream>>>(W, Wt);
    lif_gather<<<BATCH * G_SPLIT, THREADS_A, 0, stream>>>(
        Wt, ini, ids, nsp, dec, thr, states, cnt, stg);
    lif_merge<<<T_STEPS * BATCH, S_OUT, 0, stream>>>(cnt, stg, out_ids, num_out);
  } else if (ws_size >= wtBytes) {
    float* Wt = (float*)d_ws;
    wt_transpose<<<dim3(IN_DIM / 32, UNITS / 32), dim3(32, 8), 0, stream>>>(W, Wt);
    lif_monolithic<<<BATCH, THREADS_M, 0, stream>>>(Wt, 1, ini, ids, nsp, dec,
                                                    thr, out_ids, num_out, states);
  } else {
    lif_monolithic<<<BATCH, THREADS_M, 0, stream>>>(W, 0, ini, ids, nsp, dec,
                                                    thr, out_ids, num_out, states);
  }
}